// SE_21019569947046
// MI455X (gfx1250) — compile-verified
//
#include <hip/hip_runtime.h>

#define HW   12544   // 112*112
#define HW4  3136    // HW/4 (float4s per plane)
#define NB   64
#define CIN  128
#define CMID 32

typedef __attribute__((ext_vector_type(2))) float v2f;
typedef __attribute__((ext_vector_type(4))) float v4f;
typedef __attribute__((ext_vector_type(8))) float v8f;

// ---------------------------------------------------------------------------
// Pass 1: per-(b,c) spatial mean. One block per plane, 256 threads (8 waves).
// Streams x with B128 loads; regular temporal hint so the tail of x stays in
// the 192MB L2 for the scale pass.
// ---------------------------------------------------------------------------
__global__ __launch_bounds__(256) void se_mean_kernel(const float* __restrict__ x,
                                                      float* __restrict__ s) {
  const int plane = blockIdx.x;  // b*CIN + c
  const v4f* xp = (const v4f*)(x + (size_t)plane * HW);
  float acc = 0.0f;
  for (int i = threadIdx.x; i < HW4; i += 256) {
    v4f v = xp[i];
    acc += (v.x + v.y) + (v.z + v.w);
  }
  __shared__ float red[256];
  red[threadIdx.x] = acc;
  __syncthreads();
  for (int off = 128; off >= 32; off >>= 1) {
    if (threadIdx.x < (unsigned)off) red[threadIdx.x] += red[threadIdx.x + off];
    __syncthreads();
  }
  if (threadIdx.x < 32) {
    float v = red[threadIdx.x];
    #pragma unroll
    for (int off = 16; off > 0; off >>= 1) v += __shfl_down(v, off, 32);
    if (threadIdx.x == 0) s[plane] = v * (1.0f / (float)HW);
  }
}

// ---------------------------------------------------------------------------
// Pass 2: excite MLP via v_wmma_f32_16x16x4_f32 (full f32 precision).
// One workgroup, 256 threads = 8 wave32 waves, all lanes active (EXEC all-1s).
//   GEMM1: h[64x32] = relu(s[64x128] @ w1^T + b1)   -> staged in LDS
//   GEMM2: g[64x128] = hsig(h @ w2^T + b2)
// A-frag 16x4 f32: lane L<16 holds (M=L, K={k0,k0+1}); lane L>=16 (M=L-16,
// K={k0+2,k0+3}). B-frag 4x16 mirrors with N across lanes. D: M=r+8*(L>=16).
// ---------------------------------------------------------------------------
__global__ __launch_bounds__(256) void se_excite_kernel(
    const float* __restrict__ s,
    const float* __restrict__ w1, const float* __restrict__ b1,
    const float* __restrict__ w2, const float* __restrict__ b2,
    float* __restrict__ g) {
  __shared__ float h_lds[NB * CMID];  // 64 x 32
  const int tid   = threadIdx.x;
  const int wave  = tid >> 5;   // 0..7
  const int lane  = tid & 31;
  const int l16   = lane & 15;
  const int khalf = lane >> 4;  // 0 or 1 -> K pair {0,1} vs {2,3}

  // ---- GEMM1: h = relu(s @ w1^T + b1); tiles: 4 (M) x 2 (N), 1 per wave ----
  {
    const int mt  = wave >> 1;            // 0..3
    const int nt  = wave & 1;             // 0..1
    const int row = mt * 16 + l16;        // batch index for A
    const int col = nt * 16 + l16;        // mid-channel index for B
    v8f acc = {};
    for (int k0 = 0; k0 < CIN; k0 += 4) {
      const int ka = k0 + 2 * khalf;
      v2f a, bf;
      a.x  = s[row * CIN + ka];
      a.y  = s[row * CIN + ka + 1];
      bf.x = w1[col * CIN + ka];          // w1[m][c]: B^T access
      bf.y = w1[col * CIN + ka + 1];
      acc = __builtin_amdgcn_wmma_f32_16x16x4_f32(false, a, false, bf,
                                                  (short)0, acc, false, false);
    }
    const float bias = b1[col];
    #pragma unroll
    for (int r = 0; r < 8; ++r) {
      const int m = mt * 16 + r + 8 * khalf;
      h_lds[m * CMID + col] = fmaxf(acc[r] + bias, 0.0f);
    }
  }
  __syncthreads();

  // ---- GEMM2: g = hsig(h @ w2^T + b2); tiles: 4 (M) x 8 (N), 4 per wave ----
  for (int i = 0; i < 4; ++i) {
    const int t   = wave * 4 + i;  // 0..31
    const int mt  = t >> 3;        // 0..3
    const int nt  = t & 7;         // 0..7
    const int row = mt * 16 + l16;
    const int col = nt * 16 + l16; // output channel
    v8f acc = {};
    for (int k0 = 0; k0 < CMID; k0 += 4) {
      const int ka = k0 + 2 * khalf;
      v2f a, bf;
      a.x  = h_lds[row * CMID + ka];
      a.y  = h_lds[row * CMID + ka + 1];
      bf.x = w2[col * CMID + ka];        // w2[c][m]: B^T access
      bf.y = w2[col * CMID + ka + 1];
      acc = __builtin_amdgcn_wmma_f32_16x16x4_f32(false, a, false, bf,
                                                  (short)0, acc, false, false);
    }
    const float bias = b2[col];
    #pragma unroll
    for (int r = 0; r < 8; ++r) {
      const int m = mt * 16 + r + 8 * khalf;
      float v = acc[r] + bias;
      v = fminf(fmaxf(0.2f * v + 0.5f, 0.0f), 1.0f);  // hard sigmoid
      g[m * CIN + col] = v;
    }
  }
}

// ---------------------------------------------------------------------------
// Pass 3: out = x * g[plane]. Planes iterated in REVERSE so the first planes
// read are the most-recently-cached tail of x from pass 1 (~192MB L2 hits).
// Non-temporal loads/stores: the output is write-once, don't pollute L2.
// ---------------------------------------------------------------------------
__global__ __launch_bounds__(256) void se_scale_kernel(const float* __restrict__ x,
                                                       const float* __restrict__ g,
                                                       float* __restrict__ out) {
  const int plane = (int)gridDim.x - 1 - (int)blockIdx.x;
  const float gv = g[plane];
  const v4f* xp = (const v4f*)(x + (size_t)plane * HW);
  v4f* op = (v4f*)(out + (size_t)plane * HW);
  for (int i = threadIdx.x; i < HW4; i += 256) {
    v4f v = __builtin_nontemporal_load(&xp[i]);
    v *= gv;
    __builtin_nontemporal_store(v, &op[i]);
  }
}

extern "C" void kernel_launch(void* const* d_in, const int* in_sizes, int n_in,
                              void* d_out, int out_size, void* d_ws, size_t ws_size,
                              hipStream_t stream) {
  const float* x  = (const float*)d_in[0];
  const float* w1 = (const float*)d_in[1];
  const float* b1 = (const float*)d_in[2];
  const float* w2 = (const float*)d_in[3];
  const float* b2 = (const float*)d_in[4];
  float* out = (float*)d_out;

  float* s = (float*)d_ws;        // 8192 floats
  float* g = s + NB * CIN;        // 8192 floats (total 64KB of d_ws)

  se_mean_kernel<<<NB * CIN, 256, 0, stream>>>(x, s);
  se_excite_kernel<<<1, 256, 0, stream>>>(s, w1, b1, w2, b2, g);
  se_scale_kernel<<<NB * CIN, 256, 0, stream>>>(x, g, out);
}